// VariableGraphVAEEncoder_26723286515963
// MI455X (gfx1250) — compile-verified
//
#include <hip/hip_runtime.h>
#include <hip/hip_bf16.h>

typedef __attribute__((ext_vector_type(2))) float v2f;
typedef __attribute__((ext_vector_type(8))) float v8f;

#define NN 300000
#define NE 1200000
#define NG 15000
#define NODE_F 13
#define FACE_DIM 8
#define HID 64
#define HEADS 4
#define CH 16
#define LATENT 32
#define NEG_SLOPE 0.2f

// monotonic float<->uint mapping so unsigned atomicMax == float max
__device__ __forceinline__ unsigned fkey(float f) {
    unsigned u = __float_as_uint(f);
    return (u & 0x80000000u) ? ~u : (u | 0x80000000u);
}
__device__ __forceinline__ float funkey(unsigned k) {
    return __uint_as_float((k & 0x80000000u) ? (k ^ 0x80000000u) : ~k);
}
#define NEG_INF_KEY 0x007FFFFFu  // fkey(-inf)

__global__ void fill_u32(unsigned* __restrict__ p, unsigned v, long long n) {
    long long i = (long long)blockIdx.x * blockDim.x + threadIdx.x;
    if (i < n) p[i] = v;
}

// ---------------- node / edge embeddings ----------------
__global__ void embed_nodes(const float* __restrict__ x, const int* __restrict__ ft,
                            const float* __restrict__ emb, const float* __restrict__ nW,
                            const float* __restrict__ nb, float* __restrict__ h) {
    long long idx = (long long)blockIdx.x * blockDim.x + threadIdx.x;
    if (idx >= (long long)NN * HID) return;
    long long i = idx >> 6;
    int o = (int)(idx & 63);
    float acc = nb[o];
    const float* xi = x + i * NODE_F;
#pragma unroll
    for (int k = 0; k < NODE_F; ++k) acc += xi[k] * nW[k * HID + o];
    const float* em = emb + (long long)ft[i] * FACE_DIM;
#pragma unroll
    for (int k = 0; k < FACE_DIM; ++k) acc += em[k] * nW[(NODE_F + k) * HID + o];
    h[idx] = acc > 0.f ? acc : 0.f;
}

__global__ void embed_edges(const float* __restrict__ ea, const float* __restrict__ eW,
                            const float* __restrict__ eb, float* __restrict__ eh) {
    long long idx = (long long)blockIdx.x * blockDim.x + threadIdx.x;
    if (idx >= (long long)NE * HID) return;
    long long e = idx >> 6;
    int o = (int)(idx & 63);
    float acc = eb[o] + ea[e * 2 + 0] * eW[o] + ea[e * 2 + 1] * eW[HID + o];
    eh[idx] = acc > 0.f ? acc : 0.f;
}

__global__ void degree_kernel(const int* __restrict__ dst, float* __restrict__ deg) {
    long long e = (long long)blockIdx.x * blockDim.x + threadIdx.x;
    if (e >= NE) return;
    atomicAdd(&deg[dst[e]], 1.0f);
}

// ---------------- out[rows,64] = in[rows,64] @ W[64,64] via fp32 WMMA ----------------
// rows % 32 == 0. One wave -> one 16x16 tile; K=64 as 16 chained V_WMMA_F32_16X16X4_F32.
// A frag (16x4 f32): lanes 0-15 hold K=k0,k0+1 (VGPR0,1); lanes 16-31 hold K=k0+2,k0+3.
// B frag (4x16 f32): mirrored with lane%16 = N.  C/D: VGPR v holds M = v + 8*(lane/16).
__global__ __launch_bounds__(256) void gemm64_wmma(const float* __restrict__ in,
                                                   const float* __restrict__ W,
                                                   float* __restrict__ out, int rows) {
    const int lane = threadIdx.x & 31;
    const int wave = threadIdx.x >> 5;
    const int r0 = (blockIdx.x * 2 + (wave >> 2)) * 16;
    const int ntile = wave & 3;
    if (r0 >= rows) return;
    const int half = lane >> 4;
    const int lx = lane & 15;
    const long long m = r0 + lx;
    const int n = ntile * 16 + lx;
    v8f acc = {};
#pragma unroll
    for (int k0 = 0; k0 < 64; k0 += 4) {
        const int ka = k0 + 2 * half;
        v2f a, b;
        a.x = in[m * 64 + ka];
        a.y = in[m * 64 + ka + 1];
        b.x = W[ka * 64 + n];
        b.y = W[(ka + 1) * 64 + n];
        acc = __builtin_amdgcn_wmma_f32_16x16x4_f32(false, a, false, b,
                                                    (short)0, acc, false, false);
    }
#pragma unroll
    for (int v = 0; v < 8; ++v)
        out[(long long)(r0 + v + 8 * half) * 64 + ntile * 16 + lx] = acc[v];
}

// ---------------- loop_eh = segment_sum(eh, dst) / deg ----------------
__global__ void scatter_eh(const float* __restrict__ eh, const int* __restrict__ dst,
                           float* __restrict__ loop_eh) {
    long long idx = (long long)blockIdx.x * blockDim.x + threadIdx.x;
    if (idx >= (long long)NE * HID) return;
    long long e = idx >> 6;
    int c = (int)(idx & 63);
    atomicAdd(&loop_eh[(long long)dst[e] * HID + c], eh[idx]);
}

__global__ void finalize_loop_eh(float* __restrict__ loop_eh, const float* __restrict__ deg) {
    long long idx = (long long)blockIdx.x * blockDim.x + threadIdx.x;
    if (idx >= (long long)NN * HID) return;
    float d = deg[idx >> 6];
    loop_eh[idx] /= (d > 1.0f ? d : 1.0f);
}

// ---------------- attention ----------------
__global__ void node_att(const float* __restrict__ hs, const float* __restrict__ att_s,
                         const float* __restrict__ att_d, float* __restrict__ a_src,
                         float* __restrict__ a_dst) {
    long long idx = (long long)blockIdx.x * blockDim.x + threadIdx.x;
    if (idx >= (long long)NN * HEADS) return;
    long long i = idx >> 2;
    int hd = (int)(idx & 3);
    const float* f = hs + i * HID + hd * CH;
    float s0 = 0.f, s1 = 0.f;
#pragma unroll
    for (int c = 0; c < CH; ++c) {
        s0 += f[c] * att_s[hd * CH + c];
        s1 += f[c] * att_d[hd * CH + c];
    }
    a_src[idx] = s0;
    a_dst[idx] = s1;
}

// slots [0,NE) are edges, [NE, NE+NN) are self-loops (matches src_all/dst_all concat)
__global__ void attn_raw(const float* __restrict__ eh, const float* __restrict__ loop_eh,
                         const float* __restrict__ a_src, const float* __restrict__ a_dst,
                         const float* __restrict__ att_e, const int* __restrict__ src,
                         const int* __restrict__ dst, float* __restrict__ a_raw,
                         unsigned* __restrict__ mx) {
    long long idx = (long long)blockIdx.x * blockDim.x + threadIdx.x;
    if (idx >= (long long)(NE + NN) * HEADS) return;
    long long slot = idx >> 2;
    int hd = (int)(idx & 3);
    const float* f;
    long long s, d;
    if (slot < NE) { f = eh + slot * HID; s = src[slot]; d = dst[slot]; }
    else { long long i = slot - NE; f = loop_eh + i * HID; s = d = i; }
    float ae = 0.f;
#pragma unroll
    for (int c = 0; c < CH; ++c) ae += f[hd * CH + c] * att_e[hd * CH + c];
    float a = a_src[s * HEADS + hd] + a_dst[d * HEADS + hd] + ae;
    a = a > 0.f ? a : NEG_SLOPE * a;
    a_raw[idx] = a;
    atomicMax(&mx[d * HEADS + hd], fkey(a));
}

__global__ void attn_expsum(float* __restrict__ a_raw, const unsigned* __restrict__ mx,
                            float* __restrict__ ssum, const int* __restrict__ dst) {
    long long idx = (long long)blockIdx.x * blockDim.x + threadIdx.x;
    if (idx >= (long long)(NE + NN) * HEADS) return;
    long long slot = idx >> 2;
    int hd = (int)(idx & 3);
    long long d = (slot < NE) ? (long long)dst[slot] : (slot - NE);
    float e = expf(a_raw[idx] - funkey(mx[d * HEADS + hd]));
    a_raw[idx] = e;
    atomicAdd(&ssum[d * HEADS + hd], e);
}

__global__ void attn_aggregate(const float* __restrict__ hs, const float* __restrict__ a_raw,
                               const float* __restrict__ ssum, const int* __restrict__ src,
                               const int* __restrict__ dst, float* __restrict__ outb) {
    long long idx = (long long)blockIdx.x * blockDim.x + threadIdx.x;
    if (idx >= (long long)(NE + NN) * HID) return;
    long long slot = idx >> 6;
    int c = (int)(idx & 63);
    int hd = c >> 4;
    long long s, d;
    if (slot < NE) { s = src[slot]; d = dst[slot]; }
    else { s = d = slot - NE; }
    float w = a_raw[slot * HEADS + hd] / (ssum[d * HEADS + hd] + 1e-16f);
    atomicAdd(&outb[d * HID + c], hs[s * HID + c] * w);
}

__global__ void relu_bias(const float* __restrict__ outb, const float* __restrict__ bias,
                          float* __restrict__ h) {
    long long idx = (long long)blockIdx.x * blockDim.x + threadIdx.x;
    if (idx >= (long long)NN * HID) return;
    float v = outb[idx] + bias[idx & 63];
    h[idx] = v > 0.f ? v : 0.f;
}

// ---------------- pooling + VAE heads ----------------
__global__ void pool_nodes(const float* __restrict__ h, const int* __restrict__ batch,
                           const float* __restrict__ mask, float* __restrict__ hg) {
    long long idx = (long long)blockIdx.x * blockDim.x + threadIdx.x;
    if (idx >= (long long)NN * HID) return;
    long long i = idx >> 6;
    atomicAdd(&hg[(long long)batch[i] * HID + (idx & 63)], h[idx] * mask[i]);
}

__global__ void pool_cnt(const int* __restrict__ batch, const float* __restrict__ mask,
                         float* __restrict__ cnt) {
    long long i = (long long)blockIdx.x * blockDim.x + threadIdx.x;
    if (i >= NN) return;
    atomicAdd(&cnt[batch[i]], mask[i]);
}

__global__ void vae_head(const float* __restrict__ hg, const float* __restrict__ cnt,
                         const float* __restrict__ muW, const float* __restrict__ mub,
                         const float* __restrict__ lvW, const float* __restrict__ lvb,
                         float* __restrict__ out) {
    long long idx = (long long)blockIdx.x * blockDim.x + threadIdx.x;
    if (idx >= (long long)NG * LATENT) return;
    long long g = idx >> 5;
    int o = (int)(idx & 31);
    float c = cnt[g];
    float inv = 1.0f / (c > 1.0f ? c : 1.0f);
    float mu = mub[o], lv = lvb[o];
#pragma unroll
    for (int k = 0; k < HID; ++k) {
        float m = hg[g * HID + k] * inv;
        mu += m * muW[k * LATENT + o];
        lv += m * lvW[k * LATENT + o];
    }
    out[g * LATENT + o] = mu;
    out[(long long)NG * LATENT + g * LATENT + o] = lv;
}

// ---------------- host ----------------
static inline size_t align256(size_t v) { return (v + 255) & ~(size_t)255; }
#define GRID(n) ((unsigned)(((long long)(n) + 255) / 256))

extern "C" void kernel_launch(void* const* d_in, const int* in_sizes, int n_in,
                              void* d_out, int out_size, void* d_ws, size_t ws_size,
                              hipStream_t stream) {
    const float* x       = (const float*)d_in[0];
    const int*   ft      = (const int*)d_in[1];
    const int*   ei      = (const int*)d_in[2];
    const float* ea      = (const float*)d_in[3];
    const int*   batch   = (const int*)d_in[4];
    const float* mask    = (const float*)d_in[5];
    const float* emb     = (const float*)d_in[6];
    const float* nW      = (const float*)d_in[7];
    const float* nb      = (const float*)d_in[8];
    const float* eW      = (const float*)d_in[9];
    const float* eb      = (const float*)d_in[10];
    const float* gat_lin = (const float*)d_in[11];
    const float* gat_le  = (const float*)d_in[12];
    const float* att_s   = (const float*)d_in[13];
    const float* att_d   = (const float*)d_in[14];
    const float* att_e   = (const float*)d_in[15];
    const float* gbias   = (const float*)d_in[16];
    const float* muW     = (const float*)d_in[17];
    const float* mub     = (const float*)d_in[18];
    const float* lvW     = (const float*)d_in[19];
    const float* lvb     = (const float*)d_in[20];
    float* out = (float*)d_out;
    (void)in_sizes; (void)n_in; (void)out_size; (void)ws_size;

    const int* src = ei;
    const int* dst = ei + NE;

    // carve workspace
    char* base = (char*)d_ws;
    size_t off = 0;
    auto carve = [&](size_t bytes) { char* q = base + off; off = align256(off + bytes); return q; };
    float*    h       = (float*)carve((size_t)NN * HID * 4);
    float*    edge_h  = (float*)carve((size_t)NE * HID * 4);
    float*    hs      = (float*)carve((size_t)NN * HID * 4);
    float*    eh      = (float*)carve((size_t)NE * HID * 4);
    float*    loop_eh = (float*)carve((size_t)NN * HID * 4);
    float*    outb    = (float*)carve((size_t)NN * HID * 4);
    float*    deg     = (float*)carve((size_t)NN * 4);
    float*    a_src   = (float*)carve((size_t)NN * HEADS * 4);
    float*    a_dst   = (float*)carve((size_t)NN * HEADS * 4);
    float*    a_raw   = (float*)carve((size_t)(NE + NN) * HEADS * 4);
    unsigned* mx      = (unsigned*)carve((size_t)NN * HEADS * 4);
    float*    ssum    = (float*)carve((size_t)NN * HEADS * 4);
    float*    hg      = (float*)carve((size_t)NG * HID * 4);
    float*    cnt     = (float*)carve((size_t)NG * 4);

    const long long NH = (long long)NN * HID;
    const long long EH = (long long)NE * HID;
    const long long ENH4 = (long long)(NE + NN) * HEADS;
    const long long ENH64 = (long long)(NE + NN) * HID;

    embed_nodes<<<GRID(NH), 256, 0, stream>>>(x, ft, emb, nW, nb, h);
    embed_edges<<<GRID(EH), 256, 0, stream>>>(ea, eW, eb, edge_h);

    hipMemsetAsync(deg, 0, (size_t)NN * 4, stream);
    degree_kernel<<<GRID(NE), 256, 0, stream>>>(dst, deg);

    for (int l = 0; l < 3; ++l) {
        const float* Wl  = gat_lin + (size_t)l * HID * HID;
        const float* Wel = gat_le  + (size_t)l * HID * HID;
        const float* asl = att_s + (size_t)l * HEADS * CH;
        const float* adl = att_d + (size_t)l * HEADS * CH;
        const float* ael = att_e + (size_t)l * HEADS * CH;
        const float* bl  = gbias + (size_t)l * HID;

        gemm64_wmma<<<NN / 32, 256, 0, stream>>>(h, Wl, hs, NN);
        gemm64_wmma<<<NE / 32, 256, 0, stream>>>(edge_h, Wel, eh, NE);

        hipMemsetAsync(loop_eh, 0, (size_t)NN * HID * 4, stream);
        scatter_eh<<<GRID(EH), 256, 0, stream>>>(eh, dst, loop_eh);
        finalize_loop_eh<<<GRID(NH), 256, 0, stream>>>(loop_eh, deg);

        node_att<<<GRID((long long)NN * HEADS), 256, 0, stream>>>(hs, asl, adl, a_src, a_dst);
        fill_u32<<<GRID((long long)NN * HEADS), 256, 0, stream>>>(mx, NEG_INF_KEY,
                                                                  (long long)NN * HEADS);
        hipMemsetAsync(ssum, 0, (size_t)NN * HEADS * 4, stream);
        attn_raw<<<GRID(ENH4), 256, 0, stream>>>(eh, loop_eh, a_src, a_dst, ael, src, dst,
                                                 a_raw, mx);
        attn_expsum<<<GRID(ENH4), 256, 0, stream>>>(a_raw, mx, ssum, dst);

        hipMemsetAsync(outb, 0, (size_t)NN * HID * 4, stream);
        attn_aggregate<<<GRID(ENH64), 256, 0, stream>>>(hs, a_raw, ssum, src, dst, outb);
        relu_bias<<<GRID(NH), 256, 0, stream>>>(outb, bl, h);
    }

    hipMemsetAsync(hg, 0, (size_t)NG * HID * 4, stream);
    hipMemsetAsync(cnt, 0, (size_t)NG * 4, stream);
    pool_nodes<<<GRID(NH), 256, 0, stream>>>(h, batch, mask, hg);
    pool_cnt<<<GRID(NN), 256, 0, stream>>>(batch, mask, cnt);
    vae_head<<<GRID((long long)NG * LATENT), 256, 0, stream>>>(hg, cnt, muW, mub, lvW, lvb, out);
}